// CIoU_73985106641139
// MI455X (gfx1250) — compile-verified
//
#include <hip/hip_runtime.h>
#include <hip/hip_bf16.h>

typedef float v2f __attribute__((ext_vector_type(2)));
typedef float v8f __attribute__((ext_vector_type(8)));

#define NV 8          // vertices per polygon
#define MAXC 20       // clip buffer capacity (convex 8-gon ∩ 8-gon <= 16 verts)
#define INS_EPS (-1e-6f)

__device__ __forceinline__ float crossf(float ax, float ay, float bx, float by) {
    return ax * by - ay * bx;
}

// angle-sort 8 points around their centroid (Batcher 19-CE network, all static
// indices -> stays in VGPRs)
__device__ __forceinline__ void sort_poly8(float* px, float* py) {
    float cx = 0.f, cy = 0.f;
#pragma unroll
    for (int j = 0; j < NV; ++j) { cx += px[j]; cy += py[j]; }
    cx *= (1.0f / NV); cy *= (1.0f / NV);
    float ang[NV];
#pragma unroll
    for (int j = 0; j < NV; ++j) ang[j] = atan2f(py[j] - cy, px[j] - cx);
#define CSWAP(i, j)                                                     \
    if (ang[i] > ang[j]) {                                              \
        float t;                                                        \
        t = ang[i]; ang[i] = ang[j]; ang[j] = t;                        \
        t = px[i];  px[i]  = px[j];  px[j]  = t;                        \
        t = py[i];  py[i]  = py[j];  py[j]  = t;                        \
    }
    CSWAP(0,1) CSWAP(2,3) CSWAP(4,5) CSWAP(6,7)
    CSWAP(0,2) CSWAP(1,3) CSWAP(4,6) CSWAP(5,7)
    CSWAP(1,2) CSWAP(5,6)
    CSWAP(0,4) CSWAP(1,5) CSWAP(2,6) CSWAP(3,7)
    CSWAP(2,4) CSWAP(3,5)
    CSWAP(1,2) CSWAP(3,4) CSWAP(5,6)
#undef CSWAP
}

__device__ __forceinline__ float shoelace8(const float* px, const float* py) {
    float s = 0.f;
#pragma unroll
    for (int j = 0; j < NV; ++j) {
        int k = (j + 1) & 7;
        s += px[j] * py[k] - py[j] * px[k];
    }
    return 0.5f * s;
}

__global__ __launch_bounds__(256)
void ciou_compute_kernel(const float* __restrict__ A, const float* __restrict__ Bp,
                         float* __restrict__ partial, int Bsz) {
    const int i = blockIdx.x * blockDim.x + threadIdx.x;
    float ciou = 0.f;

    if (i < Bsz) {
        float pax[NV], pay[NV], pbx[NV], pby[NV];
        // (B, 8, 2) f32 -> 4 x float4 per polygon
        const float4* av = (const float4*)(A + (size_t)i * 16);
        const float4* bv = (const float4*)(Bp + (size_t)i * 16);
#pragma unroll
        for (int q = 0; q < 4; ++q) {
            float4 va = av[q];
            pax[2 * q] = va.x; pay[2 * q] = va.y; pax[2 * q + 1] = va.z; pay[2 * q + 1] = va.w;
            float4 vb = bv[q];
            pbx[2 * q] = vb.x; pby[2 * q] = vb.y; pbx[2 * q + 1] = vb.z; pby[2 * q + 1] = vb.w;
        }

        sort_poly8(pax, pay);
        sort_poly8(pbx, pby);

        const float area_a = shoelace8(pax, pay);
        const float area_b = shoelace8(pbx, pby);

        // ---- convex intersection via Sutherland–Hodgman (clip pa by pb's edges)
        float Px[MAXC], Py[MAXC], Qx[MAXC], Qy[MAXC];
        int n = NV;
#pragma unroll
        for (int j = 0; j < NV; ++j) { Px[j] = pax[j]; Py[j] = pay[j]; }

        for (int e = 0; e < NV && n > 0; ++e) {
            const float ex0 = pbx[e], ey0 = pby[e];
            const float edx = pbx[(e + 1) & 7] - ex0;
            const float edy = pby[(e + 1) & 7] - ey0;
            int m = 0;
            float sx = Px[n - 1], sy = Py[n - 1];
            float ds = edx * (sy - ey0) - edy * (sx - ex0);
            for (int k = 0; k < n; ++k) {
                const float qx = Px[k], qy = Py[k];
                const float dp = edx * (qy - ey0) - edy * (qx - ex0);
                const bool pin = (dp >= INS_EPS);
                const bool sin_ = (ds >= INS_EPS);
                if (pin != sin_) {
                    const float t = ds / (ds - dp);
                    Qx[m] = sx + t * (qx - sx);
                    Qy[m] = sy + t * (qy - sy);
                    ++m;
                }
                if (pin) { Qx[m] = qx; Qy[m] = qy; ++m; }
                sx = qx; sy = qy; ds = dp;
            }
            n = m;
            for (int k = 0; k < n; ++k) { Px[k] = Qx[k]; Py[k] = Qy[k]; }
        }

        float inter = 0.f;
        if (n >= 3) {
            float s = 0.f;
            for (int k = 0; k < n; ++k) {
                const int k2 = (k + 1 == n) ? 0 : k + 1;
                s += Px[k] * Py[k2] - Py[k] * Px[k2];
            }
            inter = fmaxf(0.5f * s, 0.f);
        }

        const float uni = area_a + area_b - inter;
        const float iou = inter / uni;

        // ---- convex hull of the 16 combined points: cross-product Jarvis march
        float hx[16], hy[16];
#pragma unroll
        for (int j = 0; j < NV; ++j) {
            hx[j] = pax[j]; hy[j] = pay[j];
            hx[j + NV] = pbx[j]; hy[j + NV] = pby[j];
        }
        int st = 0;
        for (int j = 1; j < 16; ++j)
            if (hy[j] < hy[st] || (hy[j] == hy[st] && hx[j] < hx[st])) st = j;

        int cur = st;
        float acc = 0.f;
        for (int step = 0; step < 16; ++step) {
            const float cx0 = hx[cur], cy0 = hy[cur];
            int nxt = -1;
            float nx = 0.f, ny = 0.f, nd2 = 0.f;
            for (int j = 0; j < 16; ++j) {
                const float vx = hx[j] - cx0, vy = hy[j] - cy0;
                const float d2 = vx * vx + vy * vy;
                if (d2 < 1e-16f) continue;              // skip coincident (ref: ang=inf)
                if (nxt < 0) { nxt = j; nx = vx; ny = vy; nd2 = d2; continue; }
                const float cr = crossf(nx, ny, vx, vy);
                if (cr < 0.f || (cr == 0.f && d2 > nd2)) {
                    nxt = j; nx = vx; ny = vy; nd2 = d2;  // j is more clockwise (CCW extreme)
                }
            }
            if (nxt < 0) break;
            acc += cx0 * hy[nxt] - cy0 * hx[nxt];        // includes closing edge (ref semantics)
            cur = nxt;
            if (cur == st) break;
        }
        const float ch_area = 0.5f * acc;

        ciou = iou - (ch_area - uni) / ch_area;
    }

    // deterministic block tree reduction
    __shared__ float red[256];
    red[threadIdx.x] = ciou;
    __syncthreads();
    for (int off = 128; off > 0; off >>= 1) {
        if (threadIdx.x < off) red[threadIdx.x] += red[threadIdx.x + off];
        __syncthreads();
    }
    if (threadIdx.x == 0) partial[blockIdx.x] = red[0];
}

// zero the pad region [npart, npartPadded) so the WMMA reduce can load
// unconditionally (no per-lane EXEC predication around loads)
__global__ __launch_bounds__(64)
void ciou_pad_kernel(float* __restrict__ partial, int npart, int npartPadded) {
    const int j = npart + (int)threadIdx.x;
    if (j < npartPadded) partial[j] = 0.0f;
}

// Single-wave final reduction using V_WMMA_F32_16X16X4_F32 (f32 matrix core,
// full precision). B = all-ones so D[m][n] = sum_k A[m][k] (+C): chaining C
// across chunks accumulates everything; sum(D) = 16 * total.
// Input is zero-padded to a multiple of 64 -> unconditional float2 loads,
// uniform control flow, EXEC stays all-ones throughout.
__global__ __launch_bounds__(32)
void ciou_reduce_kernel(const float* __restrict__ partial, int npartPadded,
                        float* __restrict__ out, float invB) {
    const int lane = threadIdx.x;  // exactly one wave32
    const float2* __restrict__ p2 = (const float2*)partial;

    v8f c = {};
    v2f ones; ones.x = 1.0f; ones.y = 1.0f;

    const int nchunk = npartPadded >> 6;   // 64 values per WMMA (16x4 A-tile)
    for (int ch = 0; ch < nchunk; ++ch) {
        const float2 v = p2[ch * 32 + lane];   // one global_load_b64 per lane
        v2f a; a.x = v.x; a.y = v.y;
        c = __builtin_amdgcn_wmma_f32_16x16x4_f32(
            /*neg_a=*/false, a, /*neg_b=*/false, ones,
            /*c_mod=*/(short)0, c, /*reuse_a=*/false, /*reuse_b=*/false);
    }

    float s = c[0] + c[1] + c[2] + c[3] + c[4] + c[5] + c[6] + c[7];
#pragma unroll
    for (int off = 16; off > 0; off >>= 1) s += __shfl_xor(s, off, 32);

    if (lane == 0) out[0] = s * (1.0f / 16.0f) * invB;
}

extern "C" void kernel_launch(void* const* d_in, const int* in_sizes, int n_in,
                              void* d_out, int out_size, void* d_ws, size_t ws_size,
                              hipStream_t stream) {
    const float* a = (const float*)d_in[0];
    const float* b = (const float*)d_in[1];
    float* out = (float*)d_out;
    float* partial = (float*)d_ws;

    const int Bsz = in_sizes[0] / 16;              // (B, 8, 2) f32
    const int blocks = (Bsz + 255) / 256;          // 1024 for B = 262144
    const int padded = (blocks + 63) & ~63;        // multiple of 64 for WMMA tiles

    ciou_compute_kernel<<<blocks, 256, 0, stream>>>(a, b, partial, Bsz);
    ciou_pad_kernel<<<1, 64, 0, stream>>>(partial, blocks, padded);
    ciou_reduce_kernel<<<1, 32, 0, stream>>>(partial, padded, out, 1.0f / (float)Bsz);
}